// EdgeDecoder_23871428231975
// MI455X (gfx1250) — compile-verified
//
#include <hip/hip_runtime.h>
#include <hip/hip_bf16.h>

typedef __attribute__((ext_vector_type(2))) float v2f;
typedef __attribute__((ext_vector_type(8))) float v8f;

#define N_NODES 50000
#define N_EDGES 625000
#define IN_DIMS 128

// ---------------------------------------------------------------------------
// Kernel 1: per-node projection  st[n] = { dot(x[n], W[0:128]), dot(x[n], W[128:256]) }
// One wave (32 lanes) computes a 16-node tile via V_WMMA_F32_16X16X4_F32,
// accumulating 32 K-steps of 4 to cover IN_DIMS=128. B has only columns
// N=0 (W_src) and N=1 (W_dst) nonzero; other lanes scale their load by 0
// so EXEC stays all-1s for every WMMA.
// ---------------------------------------------------------------------------
__global__ void __launch_bounds__(256)
node_proj_wmma(const float* __restrict__ x,
               const float* __restrict__ W,
               float* __restrict__ st /* [N_NODES][2] interleaved s,t */) {
    const int lane  = threadIdx.x & 31;
    const int wave  = threadIdx.x >> 5;
    const int tile  = blockIdx.x * (blockDim.x >> 5) + wave; // 16-node tile
    const int m0    = tile * 16;
    if (m0 >= N_NODES) return;   // uniform per-wave exit; 3125 full tiles exactly

    const int n = lane & 15;     // A-row within tile / B,C,D column index N
    const int g = lane >> 4;     // half-wave group: VGPR j carries K = 2g + j

    const float* xrow = x + (size_t)(m0 + n) * IN_DIMS;

    // B[k][0] = W[k] (W_src), B[k][1] = W[128+k] (W_dst), B[k][n>=2] = 0
    const float  bscale = (n < 2) ? 1.0f : 0.0f;
    const float* wrow   = W + (n < 2 ? n : 0) * IN_DIMS;

    v8f c = {};
#pragma unroll
    for (int k0 = 0; k0 < IN_DIMS; k0 += 4) {
        const int kk = k0 + 2 * g;                 // even -> 8B aligned
        v2f a  = *(const v2f*)(xrow + kk);         // A: M=n, K=kk..kk+1
        v2f bw = *(const v2f*)(wrow + kk);
        v2f b; b.x = bw.x * bscale; b.y = bw.y * bscale;
        // D = A*B + C   (8 args: neg_a, A, neg_b, B, c_mod, C, reuse_a, reuse_b)
        c = __builtin_amdgcn_wmma_f32_16x16x4_f32(false, a, false, b,
                                                  (short)0, c, false, false);
    }

    // D layout: lane L, VGPR v -> M = v + 8*(L>>4), N = L&15.
    // Columns N=0 (s) and N=1 (t) carry the results.
    if (n < 2) {
        const int mbase = m0 + 8 * g;
#pragma unroll
        for (int v = 0; v < 8; ++v) {
            st[2 * (mbase + v) + n] = c[v];
        }
    }
}

// ---------------------------------------------------------------------------
// Kernel 2: out[e] = s[src[e]] + t[dst[e]] + b    (bandwidth-bound gather)
// st table is 400 KB -> L2-resident on a 192 MB L2.
// ---------------------------------------------------------------------------
__global__ void __launch_bounds__(256)
edge_gather(const long long* __restrict__ ei,
            const float* __restrict__ st,
            const float* __restrict__ bias,
            float* __restrict__ out) {
    const int e = blockIdx.x * blockDim.x + threadIdx.x;
    if (e >= N_EDGES) return;
    const long long s = ei[e];
    const long long d = ei[(size_t)N_EDGES + e];
    out[e] = st[2 * s] + st[2 * d + 1] + bias[0];
}

extern "C" void kernel_launch(void* const* d_in, const int* in_sizes, int n_in,
                              void* d_out, int out_size, void* d_ws, size_t ws_size,
                              hipStream_t stream) {
    const float*     x  = (const float*)d_in[0];      // (50000, 128) fp32
    const long long* ei = (const long long*)d_in[1];  // (2, 625000) int64
    const float*     W  = (const float*)d_in[2];      // (1, 256) fp32
    const float*     b  = (const float*)d_in[3];      // (1,) fp32
    float*           out = (float*)d_out;             // (625000, 1) fp32
    float*           st  = (float*)d_ws;              // 50000 * 2 floats = 400 KB

    const int tiles   = (N_NODES + 15) / 16;          // 3125 (exact)
    const int wavesPB = 256 / 32;                     // 8 waves per block
    const int grid1   = (tiles + wavesPB - 1) / wavesPB;
    node_proj_wmma<<<grid1, 256, 0, stream>>>(x, W, st);

    const int grid2 = (N_EDGES + 255) / 256;
    edge_gather<<<grid2, 256, 0, stream>>>(ei, st, b, out);
}